// SAGCN_52432960749947
// MI455X (gfx1250) — compile-verified
//
#include <hip/hip_runtime.h>

#define WAVE 32
#define DCOLS 64

// ---- gfx1250 async-to-LDS path (probe-verified: 4 args, generic int*) ------
#if defined(__gfx1250__) && __has_builtin(__builtin_amdgcn_global_load_async_to_lds_b32)
#define HAVE_ASYNC_LDS 1
__device__ __forceinline__ void wait_asynccnt0() {
#if __has_builtin(__builtin_amdgcn_s_wait_asynccnt)
  __builtin_amdgcn_s_wait_asynccnt(0);
#else
  asm volatile("s_wait_asynccnt 0" ::: "memory");
#endif
}
#endif

// ---------------------------------------------------------------------------
// out[i] = 0.25 * (w0*a + w1*b + w2*c)   (layer-0 term of all three factors)
// a/b/c are read exactly once here -> non-temporal loads keep L2 for the
// hot SpMM working set.
__global__ __launch_bounds__(256) void k_init_out(
    float* __restrict__ out, const float* __restrict__ a,
    const float* __restrict__ b, const float* __restrict__ c,
    const float* __restrict__ w, int n) {
  int i = blockIdx.x * 256 + threadIdx.x;
  if (i < n) {
    float va = __builtin_nontemporal_load(a + i);
    float vb = __builtin_nontemporal_load(b + i);
    float vc = __builtin_nontemporal_load(c + i);
    out[i] = 0.25f * (w[0] * va + w[1] * vb + w[2] * vc);
  }
}

__global__ __launch_bounds__(256) void k_zero_i32(int* __restrict__ p, int n) {
  int i = blockIdx.x * 256 + threadIdx.x;
  if (i < n) p[i] = 0;
}

// histogram of row indices (rows streamed once -> NT load)
__global__ __launch_bounds__(256) void k_hist(
    const int* __restrict__ rows, int* __restrict__ cnt, int nnz) {
  int i = blockIdx.x * 256 + threadIdx.x;
  if (i < nnz) {
    int r = __builtin_nontemporal_load(rows + i);
    atomicAdd(&cnt[r], 1);
  }
}

// per-block exclusive scan (256 elems), stages input through LDS via the
// gfx1250 async global->LDS copy.
__global__ __launch_bounds__(256) void k_scan_block(
    const int* __restrict__ counts, int* __restrict__ scanned,
    int* __restrict__ bsums, int n) {
  __shared__ int sh[256];
  const int t = threadIdx.x;
  const int i = blockIdx.x * 256 + t;
#ifdef HAVE_ASYNC_LDS
  if (i < n) {
    __builtin_amdgcn_global_load_async_to_lds_b32(
        (int*)(counts + i), (int*)&sh[t], 0, 0);
  } else {
    sh[t] = 0;
  }
  wait_asynccnt0();
  __syncthreads();
  const int v = sh[t];
#else
  const int v = (i < n) ? counts[i] : 0;
  sh[t] = v;
  __syncthreads();
#endif
  for (int off = 1; off < 256; off <<= 1) {
    int add = (t >= off) ? sh[t - off] : 0;
    __syncthreads();
    sh[t] += add;
    __syncthreads();
  }
  const int incl = sh[t];
  if (i < n) scanned[i] = incl - v;       // exclusive
  if (t == 255) bsums[blockIdx.x] = incl; // block total
}

// single-block exclusive scan of block sums (nb <= 1024)
__global__ __launch_bounds__(1024) void k_scan_sums(int* __restrict__ s, int nb) {
  __shared__ int sh[1024];
  const int t = threadIdx.x;
  const int v = (t < nb) ? s[t] : 0;
  sh[t] = v;
  __syncthreads();
  for (int off = 1; off < 1024; off <<= 1) {
    int add = (t >= off) ? sh[t - off] : 0;
    __syncthreads();
    sh[t] += add;
    __syncthreads();
  }
  if (t < nb) s[t] = sh[t] - v;
}

// row_ptr[i] += bsums[block]; cursor[i] = row_ptr[i]; row_ptr[n] = nnz
__global__ __launch_bounds__(256) void k_scan_finalize(
    int* __restrict__ row_ptr, const int* __restrict__ bsums,
    int* __restrict__ cursor, int n, int nnz, int* __restrict__ tail) {
  int i = blockIdx.x * 256 + threadIdx.x;
  if (i < n) {
    int v = row_ptr[i] + bsums[blockIdx.x];
    row_ptr[i] = v;
    cursor[i] = v;
  }
  if (i == 0) *tail = nnz;
}

// bucket (counting-sort) scatter into CSR entry array: entry = {col, val}
// rows/cols/vals streamed once -> NT loads.
__global__ __launch_bounds__(256) void k_scatter(
    const int* __restrict__ rows, const int* __restrict__ cols,
    const float* __restrict__ vals, int* __restrict__ cursor,
    int2* __restrict__ entries, int nnz) {
  int i = blockIdx.x * 256 + threadIdx.x;
  if (i < nnz) {
    int r = __builtin_nontemporal_load(rows + i);
    int c = __builtin_nontemporal_load(cols + i);
    float v = __builtin_nontemporal_load(vals + i);
    int p = atomicAdd(&cursor[r], 1);
    entries[p] = make_int2(c, __float_as_int(v));
  }
}

// Pull-mode CSR SpMM fused with the weighted output accumulation.
// Wave32 per row, 2 f32 columns per lane, zero atomics:
//   y[row]   = A[row,:] @ x          (skipped when store_y == 0)
//   out[row] += 0.25*w[f] * (A[row,:] @ x)
__global__ __launch_bounds__(256) void k_spmm_fused(
    const int* __restrict__ row_ptr, const int2* __restrict__ entries,
    const float* __restrict__ x, float* __restrict__ y,
    float* __restrict__ out, const float* __restrict__ w, int f, int store_y,
    int n) {
  const int wave = threadIdx.x >> 5;
  const int lane = threadIdx.x & 31;
  const int row = blockIdx.x * 8 + wave;
  if (row >= n) return;
  const int start = row_ptr[row];
  const int end = row_ptr[row + 1];
  float acc0 = 0.0f, acc1 = 0.0f;
  const int coff = lane << 1;
  int j = start;
  for (; j + 3 < end; j += 4) {
    const int2 e0 = entries[j];
    const int2 e1 = entries[j + 1];
    const int2 e2 = entries[j + 2];
    const int2 e3 = entries[j + 3];
    const float2 x0 = *(const float2*)(x + ((size_t)e0.x << 6) + coff);
    const float2 x1 = *(const float2*)(x + ((size_t)e1.x << 6) + coff);
    const float2 x2 = *(const float2*)(x + ((size_t)e2.x << 6) + coff);
    const float2 x3 = *(const float2*)(x + ((size_t)e3.x << 6) + coff);
    const float v0 = __int_as_float(e0.y);
    const float v1 = __int_as_float(e1.y);
    const float v2 = __int_as_float(e2.y);
    const float v3 = __int_as_float(e3.y);
    acc0 = fmaf(v0, x0.x, acc0); acc1 = fmaf(v0, x0.y, acc1);
    acc0 = fmaf(v1, x1.x, acc0); acc1 = fmaf(v1, x1.y, acc1);
    acc0 = fmaf(v2, x2.x, acc0); acc1 = fmaf(v2, x2.y, acc1);
    acc0 = fmaf(v3, x3.x, acc0); acc1 = fmaf(v3, x3.y, acc1);
  }
  for (; j < end; ++j) {
    const int2 e = entries[j];
    const float v = __int_as_float(e.y);
    const float2 xv = *(const float2*)(x + ((size_t)e.x << 6) + coff);
    acc0 = fmaf(v, xv.x, acc0);
    acc1 = fmaf(v, xv.y, acc1);
  }
  const size_t ro = ((size_t)row << 6) + coff;
  if (store_y) {
    *(float2*)(y + ro) = make_float2(acc0, acc1);
  }
  const float s = 0.25f * w[f];
  float2 o = *(const float2*)(out + ro);
  o.x = fmaf(s, acc0, o.x);
  o.y = fmaf(s, acc1, o.y);
  *(float2*)(out + ro) = o;
}

// ---------------------------------------------------------------------------
extern "C" void kernel_launch(void* const* d_in, const int* in_sizes, int n_in,
                              void* d_out, int out_size, void* d_ws,
                              size_t ws_size, hipStream_t stream) {
  (void)n_in; (void)out_size; (void)ws_size;
  const float* emb[3]  = {(const float*)d_in[0], (const float*)d_in[1],
                          (const float*)d_in[2]};
  const int*   gidx[3] = {(const int*)d_in[3], (const int*)d_in[5],
                          (const int*)d_in[7]};
  const float* gval[3] = {(const float*)d_in[4], (const float*)d_in[6],
                          (const float*)d_in[8]};
  const float* w = (const float*)d_in[9];

  const int nnz = in_sizes[4];        // 4,800,000
  const int nd  = in_sizes[0];        // N * 64
  const int n   = nd / DCOLS;         // 150,000

  // workspace carve-out
  char* ws = (char*)d_ws;
  size_t off = 0;
  auto carve = [&](size_t bytes) -> void* {
    void* p = ws + off;
    off += (bytes + 255) & ~(size_t)255;
    return p;
  };
  int2*  entries = (int2*) carve((size_t)nnz * sizeof(int2));
  float* cur     = (float*)carve((size_t)nd * sizeof(float));
  float* nxt     = (float*)carve((size_t)nd * sizeof(float));
  int*   row_ptr = (int*)  carve((size_t)(n + 1) * sizeof(int));
  int*   cursor  = (int*)  carve((size_t)n * sizeof(int));
  int*   bsums   = (int*)  carve(4096);

  const int gb_nd   = (nd + 255) / 256;
  const int gb_nnz  = (nnz + 255) / 256;
  const int gb_n    = (n + 255) / 256;   // also #scan blocks (<= 1024)
  const int gb_rows = (n + 7) / 8;

  float* out = (float*)d_out;
  // layer-0 contribution of all three factors
  k_init_out<<<gb_nd, 256, 0, stream>>>(out, emb[0], emb[1], emb[2], w, nd);

  for (int f = 0; f < 3; ++f) {
    const int* rows = gidx[f];
    const int* cols = gidx[f] + nnz;

    // build CSR for this factor (reused across its 3 layers)
    k_zero_i32     <<<gb_n,   256, 0, stream>>>(cursor, n);
    k_hist         <<<gb_nnz, 256, 0, stream>>>(rows, cursor, nnz);
    k_scan_block   <<<gb_n,   256, 0, stream>>>(cursor, row_ptr, bsums, n);
    k_scan_sums    <<<1,     1024, 0, stream>>>(bsums, gb_n);
    k_scan_finalize<<<gb_n,   256, 0, stream>>>(row_ptr, bsums, cursor, n, nnz,
                                                row_ptr + n);
    k_scatter      <<<gb_nnz, 256, 0, stream>>>(rows, cols, gval[f], cursor,
                                                entries, nnz);

    // 3 propagation layers, accumulation fused into the SpMM:
    //   e_{k+1} = A e_k ;  out += 0.25*w[f]*e_{k+1}
    const float* xsrc = emb[f];
    float* ydst = cur;
    for (int layer = 0; layer < 3; ++layer) {
      const int store_y = (layer < 2) ? 1 : 0;  // last layer feeds only `out`
      k_spmm_fused<<<gb_rows, 256, 0, stream>>>(row_ptr, entries, xsrc, ydst,
                                                out, w, f, store_y, n);
      xsrc = ydst;
      ydst = (ydst == cur) ? nxt : cur;
    }
  }
}